// EdgeGCN_55078660604120
// MI455X (gfx1250) — compile-verified
//
#include <hip/hip_runtime.h>

// EdgeGCN for MI455X (gfx1250, wave32). fp32 WMMA 16x16x4 everywhere.
// Algebraic refactor: edge MLP concat-GEMM is split into per-node P1/P2
// precomputation + tiny K=32 per-edge GEMM -> memory-bound (~2.3GB traffic).

typedef __attribute__((ext_vector_type(2))) float v2f;
typedef __attribute__((ext_vector_type(8))) float v8f;

#define D   128   // node / hidden dim
#define DE  32    // edge feature dim

__device__ __forceinline__ v8f wmma4(float a0, float a1, float b0, float b1, v8f c) {
  v2f a; a[0] = a0; a[1] = a1;
  v2f b; b[0] = b0; b[1] = b1;
  // (neg_a, A, neg_b, B, c_mod, C, reuse_a, reuse_b)
  return __builtin_amdgcn_wmma_f32_16x16x4_f32(false, a, false, b, (short)0, c,
                                               false, false);
}

// ---------------------------------------------------------------- utilities
__global__ void zero_f32(float* __restrict__ p, long n) {
  long i = (long)blockIdx.x * blockDim.x + threadIdx.x;
  long stride = (long)gridDim.x * blockDim.x;
  for (; i < n; i += stride) p[i] = 0.0f;
}

// One wave (32 lanes) per edge: lane handles 4 contiguous floats of the row.
__global__ void scatter_add(const float* __restrict__ X, const int* __restrict__ src,
                            const int* __restrict__ dst, float* __restrict__ sum,
                            float* __restrict__ cnt, int E) {
  const int wave = threadIdx.x >> 5;
  const int lane = threadIdx.x & 31;
  const int e = blockIdx.x * 8 + wave;
  if (e >= E) return;
  const int s = src[e];
  const int d = dst[e];
  const float4 v = *(const float4*)(X + (long)s * D + lane * 4);
  float* o = sum + (long)d * D + lane * 4;
  atomicAdd(o + 0, v.x);
  atomicAdd(o + 1, v.y);
  atomicAdd(o + 2, v.z);
  atomicAdd(o + 3, v.w);
  if (lane == 0) atomicAdd(cnt + d, 1.0f);
}

// sum[i] /= max(cnt[row],1)  (in place -> mean aggregate)
__global__ void finalize_mean(float* __restrict__ sum, const float* __restrict__ cnt,
                              long n) {
  long i = (long)blockIdx.x * blockDim.x + threadIdx.x;
  long stride = (long)gridDim.x * blockDim.x;
  for (; i < n; i += stride) {
    float c = cnt[i >> 7];            // D == 128
    sum[i] = sum[i] / fmaxf(c, 1.0f);
  }
}

// ---------------------------------------------------- fused node-level GEMM
// Hout[M,128] = act( A1[M,128] @ W1[128,128] (+ A2 @ W2) (+ bias) )
// Block = 256 threads = 8 waves; tile = 16 rows x 128 cols; wave w -> 16 cols.
// Requires M % 16 == 0 (N_NODES = 50000 = 3125*16).
__global__ void node_gemm(const float* __restrict__ A1, const float* __restrict__ W1,
                          const float* __restrict__ A2, const float* __restrict__ W2,
                          const float* __restrict__ bias, float* __restrict__ Hout,
                          int do_relu) {
  const int tid  = threadIdx.x;
  const int wave = tid >> 5;
  const int lane = tid & 31;
  const int half = lane >> 4;     // 0: K={0,1}, 1: K={2,3} (fp32 A/B layout)
  const int lm   = lane & 15;
  const int m0   = blockIdx.x * 16;
  const int n    = wave * 16 + lm;

  v8f c;
  const float bv = bias ? bias[n] : 0.0f;
#pragma unroll
  for (int r = 0; r < 8; ++r) c[r] = bv;

  // A fragment: lane -> row m0+lm, K pair {k0+2*half, k0+2*half+1} (float2)
  const float* aRow = A1 + (long)(m0 + lm) * D + 2 * half;
#pragma unroll 4
  for (int k0 = 0; k0 < D; k0 += 4) {
    const float2 a = *(const float2*)(aRow + k0);
    const float b0 = W1[(k0 + 2 * half) * D + n];
    const float b1 = W1[(k0 + 2 * half + 1) * D + n];
    c = wmma4(a.x, a.y, b0, b1, c);
  }
  if (A2) {
    const float* aRow2 = A2 + (long)(m0 + lm) * D + 2 * half;
#pragma unroll 4
    for (int k0 = 0; k0 < D; k0 += 4) {
      const float2 a = *(const float2*)(aRow2 + k0);
      const float b0 = W2[(k0 + 2 * half) * D + n];
      const float b1 = W2[(k0 + 2 * half + 1) * D + n];
      c = wmma4(a.x, a.y, b0, b1, c);
    }
  }

  // C layout: VGPR r -> row r (lanes 0-15) / row r+8 (lanes 16-31), col n.
#pragma unroll
  for (int r = 0; r < 8; ++r) {
    float v = c[r];
    if (do_relu) v = fmaxf(v, 0.0f);
    const int m = m0 + r + 8 * half;
    Hout[(long)m * D + n] = v;
  }
}

// ------------------------------------------------------------- edge scoring
// out[e] = relu( P1[src] + P2[dst] + ea @ W3 + bc1 ) . Wc2 + bc2
// Block = 256 threads = 8 waves; 16 edges per block; wave w -> hid cols 16w..16w+15.
__global__ void edge_mlp(const float* __restrict__ P1, const float* __restrict__ P2,
                         const float* __restrict__ ea, const int* __restrict__ src,
                         const int* __restrict__ dst, const float* __restrict__ W3,
                         const float* __restrict__ bc1, const float* __restrict__ Wc2,
                         const float* __restrict__ bc2, float* __restrict__ out,
                         int E) {
  __shared__ int   s_src[16];
  __shared__ int   s_dst[16];
  __shared__ float s_red[8][16];

  const int tid  = threadIdx.x;
  const int wave = tid >> 5;
  const int lane = tid & 31;
  const int half = lane >> 4;
  const int lm   = lane & 15;
  const int e0   = blockIdx.x * 16;
  const int n    = wave * 16 + lm;

  if (tid < 16)       s_src[tid]      = src[e0 + tid];
  else if (tid < 32)  s_dst[tid - 16] = dst[e0 + tid - 16];
  __syncthreads();

  // Seed the WMMA accumulator with the gathered node terms + bias.
  v8f c;
  const float bv = bc1[n];
#pragma unroll
  for (int r = 0; r < 8; ++r) {
    const int m = r + 8 * half;
    c[r] = P1[(long)s_src[m] * D + n] + P2[(long)s_dst[m] * D + n] + bv;
  }

  // Accumulate ea[16 x 32] @ W3[32 x 128] over 8 WMMA k-steps.
  const float* aRow = ea + (long)(e0 + lm) * DE + 2 * half;
#pragma unroll
  for (int k0 = 0; k0 < DE; k0 += 4) {
    const float2 a = *(const float2*)(aRow + k0);
    const float b0 = W3[(k0 + 2 * half) * D + n];
    const float b1 = W3[(k0 + 2 * half + 1) * D + n];
    c = wmma4(a.x, a.y, b0, b1, c);
  }

  // relu -> dot with Wc2 segment -> reduce 16 lanes (half-wave) per row.
  const float w2 = Wc2[n];
#pragma unroll
  for (int r = 0; r < 8; ++r) {
    float t = fmaxf(c[r], 0.0f) * w2;
    t += __shfl_xor(t, 1);
    t += __shfl_xor(t, 2);
    t += __shfl_xor(t, 4);
    t += __shfl_xor(t, 8);
    if (lm == 0) s_red[wave][r + 8 * half] = t;
  }
  __syncthreads();

  if (tid < 16) {
    float acc = bc2[0];
#pragma unroll
    for (int w = 0; w < 8; ++w) acc += s_red[w][tid];
    if (e0 + tid < E) out[e0 + tid] = acc;
  }
}

// ------------------------------------------------------------------- driver
extern "C" void kernel_launch(void* const* d_in, const int* in_sizes, int n_in,
                              void* d_out, int out_size, void* d_ws, size_t ws_size,
                              hipStream_t stream) {
  (void)n_in; (void)out_size; (void)ws_size;
  const float* x   = (const float*)d_in[0];
  const float* ea  = (const float*)d_in[1];
  const float* W1a = (const float*)d_in[2];
  const float* b1  = (const float*)d_in[3];
  const float* W1r = (const float*)d_in[4];
  const float* W2a = (const float*)d_in[5];
  const float* b2  = (const float*)d_in[6];
  const float* W2r = (const float*)d_in[7];
  const float* Wc1 = (const float*)d_in[8];   // [288,128] row-major
  const float* bc1 = (const float*)d_in[9];
  const float* Wc2 = (const float*)d_in[10];  // [128,1]
  const float* bc2 = (const float*)d_in[11];
  const int*   ei  = (const int*)d_in[12];    // [2,E]

  const int N = in_sizes[0] / D;              // 50000 (divisible by 16)
  const int E = in_sizes[12] / 2;             // 800000 (divisible by 16)
  const int* src = ei;
  const int* dst = ei + E;

  float* ws = (float*)d_ws;
  const long NM = (long)N * D;
  float* bufA = ws;            // sum/agg, later P2
  float* bufB = ws + NM;       // h1, later P1
  float* bufC = ws + 2 * NM;   // h2
  float* cnt  = ws + 3 * NM;   // [N] in-degree
  float* out  = (float*)d_out;

  const dim3 blk(256);
  const int sgrid = (E + 7) / 8;
  const int ggrid = N / 16;
  const int egrid = E / 16;

  // ---- SAGE layer 1: h1 = relu(mean_agg(x) @ W1a + x @ W1r + b1) ----
  zero_f32<<<2048, blk, 0, stream>>>(bufA, NM);
  zero_f32<<<128,  blk, 0, stream>>>(cnt, N);
  scatter_add<<<sgrid, blk, 0, stream>>>(x, src, dst, bufA, cnt, E);
  finalize_mean<<<4096, blk, 0, stream>>>(bufA, cnt, NM);
  node_gemm<<<ggrid, blk, 0, stream>>>(bufA, W1a, x, W1r, b1, bufB, 1);

  // ---- SAGE layer 2: h2 = mean_agg(h1) @ W2a + h1 @ W2r + b2 ----
  zero_f32<<<2048, blk, 0, stream>>>(bufA, NM);
  zero_f32<<<128,  blk, 0, stream>>>(cnt, N);
  scatter_add<<<sgrid, blk, 0, stream>>>(bufB, src, dst, bufA, cnt, E);
  finalize_mean<<<4096, blk, 0, stream>>>(bufA, cnt, NM);
  node_gemm<<<ggrid, blk, 0, stream>>>(bufA, W2a, bufB, W2r, b2, bufC, 0);

  // ---- Factor the edge concat-GEMM into per-node precomputes ----
  // P1 = h2 @ Wc1[0:128,:]   (src term)   -> bufB (h1 slot, now free)
  // P2 = h2 @ Wc1[128:256,:] (dst term)   -> bufA (agg slot, now free)
  node_gemm<<<ggrid, blk, 0, stream>>>(bufC, Wc1,           nullptr, nullptr,
                                       nullptr, bufB, 0);
  node_gemm<<<ggrid, blk, 0, stream>>>(bufC, Wc1 + 128 * D, nullptr, nullptr,
                                       nullptr, bufA, 0);

  // ---- Edge MLP: out = relu(P1[src]+P2[dst]+ea@Wc1[256:288]+bc1).Wc2 + bc2
  edge_mlp<<<egrid, blk, 0, stream>>>(bufB, bufA, ea, src, dst, Wc1 + 256 * D,
                                      bc1, Wc2, bc2, out, E);
}